// AU_54606214201637
// MI455X (gfx1250) — compile-verified
//
#include <hip/hip_runtime.h>

typedef __attribute__((ext_vector_type(2))) float v2f;
typedef __attribute__((ext_vector_type(4))) float v4f;
typedef __attribute__((ext_vector_type(8))) float v8f;

#define BB 8
#define CC 128
#define LL 2048
#define PITCH 132   // feat row pitch: 528 B (16B-aligned rows, (4*dl+c)%64 banks)
#define LOG2E 1.44269504088896340736f

#if !__has_builtin(__builtin_amdgcn_exp2f)
#define __builtin_amdgcn_exp2f exp2f
#endif

// ---------------------------------------------------------------------------
// Kernel 1: pooled_pad[16][128]; rows 0..7 = mean over L of x[b,c,:], rows
// 8..15 = 0 (zero padding so the WMMA A-matrix needs no EXEC masking).
// grid = 2048 blocks (one element each), 256 threads.
// ---------------------------------------------------------------------------
__global__ void AU_pool_kernel(const float* __restrict__ x,
                               float* __restrict__ pooled_pad) {
    __shared__ float red[256];
    const int row = blockIdx.x;          // 0..2047
    const int tid = threadIdx.x;
    if (row < BB * CC) {                 // real rows: row = b*128 + c
        const float* p = x + (size_t)row * LL;
        float s = 0.f;
        for (int i = tid; i < LL; i += 256) s += p[i];
        red[tid] = s;
        __syncthreads();
        for (int off = 128; off > 0; off >>= 1) {
            if (tid < off) red[tid] += red[tid + off];
            __syncthreads();
        }
        if (tid == 0) pooled_pad[row] = red[0] * (1.0f / (float)LL);
    } else {
        if (tid == 0) pooled_pad[row] = 0.f;   // padding rows 8..15
    }
}

// ---------------------------------------------------------------------------
// Kernel 2: encode[8][128] = pooled @ W^T + bias via V_WMMA_F32_16X16X4_F32.
// One block, 8 waves; wave w computes output N-tile [16w, 16w+16), M=16
// (rows 8..15 are padding), K=128 as 32 chained k=4 WMMAs.
// A 16x4 layout: lanes 0-15 hold M=lane {K0,K1}; lanes 16-31 hold {K2,K3}.
// B 4x16 layout: lanes 0-15 hold N=lane rows {K0,K1}; lanes 16-31 {K2,K3}.
// D 16x16: VGPR v -> M = v (+8 for lanes 16-31), N = lane%16.
// ---------------------------------------------------------------------------
__global__ void AU_encode_kernel(const float* __restrict__ pooled_pad,
                                 const float* __restrict__ W,
                                 const float* __restrict__ bias,
                                 float* __restrict__ encode) {
    const int lane  = threadIdx.x & 31;
    const int wtile = threadIdx.x >> 5;      // 0..7 N-tile
    const int half  = lane >> 4;             // 0: K0/K1, 1: K2/K3
    const int lid   = lane & 15;
    const int n     = wtile * 16 + lid;

    v8f acc = {};
    for (int kk = 0; kk < CC; kk += 4) {
        const int k0 = kk + half * 2;
        // A = pooled_pad (M=lid, K=k0..k0+1); 8-byte aligned pair load
        v2f a = *(const v2f*)&pooled_pad[lid * CC + k0];
        // B[k][n] = W[n][k]  (encode = pooled @ W^T)
        v2f bm = *(const v2f*)&W[n * CC + k0];
        acc = __builtin_amdgcn_wmma_f32_16x16x4_f32(
            false, a, false, bm, (short)0, acc, false, false);
    }
#pragma unroll
    for (int v = 0; v < 8; ++v) {
        const int m = half ? (v + 8) : v;
        if (m < BB) encode[m * CC + n] = acc[v] + bias[n];
    }
}

// ---------------------------------------------------------------------------
// Kernel 3: main per-token softmax-weighted reduction.
//   aug[t,j] = sum_k exp(f_j * e_k) * f_k / sum_k exp(f_j * e_k)
// One workgroup = 32 consecutive tokens (same batch). 8 waves x 4 tokens.
// Per lane: 4 j-rows; each k-step-of-4 issues 2 ds_load_b128 + 16 exp
// (TRANS) + 48 VALU -> good TRANS/VALU co-execution, minimal DS issue load.
// Softmax max-shift folded into the fma (arg = a*es_k - base, es = e*log2e).
// ---------------------------------------------------------------------------
__global__ void AU_attn_kernel(const float* __restrict__ x,
                               const float* __restrict__ encode,
                               float* __restrict__ out) {
    __shared__ __align__(16) float feat[32 * PITCH];  // [token][c]
    __shared__ __align__(16) float es[BB * CC];       // encode * log2(e)
    __shared__ float emax[BB], emin[BB];

    const int tid = threadIdx.x;
    const int t0  = blockIdx.x * 32;       // first token of this block
    const int b   = t0 / LL;
    const int l0  = t0 % LL;               // multiple of 32
    const float* xb = x + (size_t)b * CC * LL;

    // stage encode * log2e
    for (int i = tid; i < BB * CC; i += 256) es[i] = encode[i] * LOG2E;

    // stage feat: coalesced 128B global loads
    const int dl = tid & 31;
    const int cb = tid >> 5;
#pragma unroll
    for (int p = 0; p < 16; ++p) {
        const int c = cb + p * 8;
        feat[dl * PITCH + c] = xb[(size_t)c * LL + l0 + dl];
    }
    __syncthreads();

    if (tid < BB) {
        float mx = -1e30f, mn = 1e30f;
        for (int k = 0; k < CC; ++k) {
            const float v = es[tid * CC + k];
            mx = fmaxf(mx, v);
            mn = fminf(mn, v);
        }
        emax[tid] = mx;
        emin[tid] = mn;
    }
    __syncthreads();

    const int lane = tid & 31;
    const int wv   = tid >> 5;
    for (int ti = 0; ti < 4; ++ti) {
        const int tok = wv * 4 + ti;
        const int r   = tok & 7;           // (l0 + tok) % 8 == tok % 8
        float* fr = &feat[tok * PITCH];
        const float* er = &es[r * CC];

        const float a0 = fr[lane];
        const float a1 = fr[lane + 32];
        const float a2 = fr[lane + 64];
        const float a3 = fr[lane + 96];
        const float mx = emax[r], mn = emin[r];
        const float s0 = a0 * (a0 >= 0.f ? mx : mn);
        const float s1 = a1 * (a1 >= 0.f ? mx : mn);
        const float s2 = a2 * (a2 >= 0.f ? mx : mn);
        const float s3 = a3 * (a3 >= 0.f ? mx : mn);

        float n0 = 0.f, n1 = 0.f, n2 = 0.f, n3 = 0.f;
        float d0 = 0.f, d1 = 0.f, d2 = 0.f, d3 = 0.f;
        for (int k = 0; k < CC; k += 4) {
            const v4f ev = *(const v4f*)&er[k];   // ds_load_b128 broadcast
            const v4f fv = *(const v4f*)&fr[k];   // ds_load_b128 broadcast
#pragma unroll
            for (int u = 0; u < 4; ++u) {
                const float ek = ev[u];
                const float fk = fv[u];
                const float e0 = __builtin_amdgcn_exp2f(fmaf(a0, ek, -s0));
                const float e1 = __builtin_amdgcn_exp2f(fmaf(a1, ek, -s1));
                const float e2 = __builtin_amdgcn_exp2f(fmaf(a2, ek, -s2));
                const float e3 = __builtin_amdgcn_exp2f(fmaf(a3, ek, -s3));
                n0 = fmaf(e0, fk, n0); d0 += e0;
                n1 = fmaf(e1, fk, n1); d1 += e1;
                n2 = fmaf(e2, fk, n2); d2 += e2;
                n3 = fmaf(e3, fk, n3); d3 += e3;
            }
        }
        // all reads of fr[] done for this wave's token -> overwrite with aug
        fr[lane]      = n0 / d0;
        fr[lane + 32] = n1 / d1;
        fr[lane + 64] = n2 / d2;
        fr[lane + 96] = n3 / d3;
    }
    __syncthreads();

    // coalesced 128B stores: out[b, c, l0+dl] = aug[dl][c]
    float* ob = out + (size_t)b * CC * LL;
#pragma unroll
    for (int p = 0; p < 16; ++p) {
        const int c = cb + p * 8;
        ob[(size_t)c * LL + l0 + dl] = feat[dl * PITCH + c];
    }
}

// ---------------------------------------------------------------------------
extern "C" void kernel_launch(void* const* d_in, const int* in_sizes, int n_in,
                              void* d_out, int out_size, void* d_ws, size_t ws_size,
                              hipStream_t stream) {
    const float* x  = (const float*)d_in[0];   // (8, 128, 2048)
    const float* W  = (const float*)d_in[1];   // (128, 128)
    const float* bb = (const float*)d_in[2];   // (128,)
    float* out = (float*)d_out;                // (8, 128, 2048)

    float* pooled_pad = (float*)d_ws;          // 16*128 floats (rows 8..15 = 0)
    float* encode     = pooled_pad + 16 * CC;  // 8*128 floats

    AU_pool_kernel<<<16 * CC, 256, 0, stream>>>(x, pooled_pad);
    AU_encode_kernel<<<1, 256, 0, stream>>>(pooled_pad, W, bb, encode);
    AU_attn_kernel<<<BB * LL / 32, 256, 0, stream>>>(x, encode, out);
}